// NeighborAwareLoss_76347338654269
// MI455X (gfx1250) — compile-verified
//
#include <hip/hip_runtime.h>

typedef __attribute__((ext_vector_type(2))) float v2f;
typedef __attribute__((ext_vector_type(8))) float v8f;

#define B_ROWS  16384
#define D_DST   511
#define T_TUN   4088      // = 8 * 511, divisible by 4 -> 1022 K-chunks
#define L_LINKS 16
#define EPS     1e-8f

// ---------------------------------------------------------------------------
// Stage 1: link_traffic[B,16] = sum_t pred[b,t]*demands[b,t>>3] * onehot(ttl[t])
// Done as a GEMM with V_WMMA_F32_16X16X4_F32: A = 16x4 tunnel_traffic tile,
// B = 4x16 one-hot link-selection matrix (exact f32 math).
//
// A layout (ISA 32-bit A 16x4): lanes 0-15 = rows M, VGPR0=K0, VGPR1=K1;
//                               lanes 16-31 = rows M, VGPR0=K2, VGPR1=K3.
// B layout mirrors with lanes as N columns.
// C layout: VGPR i -> (M=i, N=lane) for lanes 0-15, (M=8+i, N=lane-16) for 16-31.
// ---------------------------------------------------------------------------
__global__ __launch_bounds__(128)
void nal_stage1_link_traffic(const float* __restrict__ pred,
                             const float* __restrict__ demands,
                             const int*   __restrict__ ttl,
                             float*       __restrict__ link_traffic) {
    __shared__ float red[4 * 256];           // 4 waves x (32 lanes x 8 floats)
    const int tid   = threadIdx.x;
    const int wave  = tid >> 5;
    const int lane  = tid & 31;
    const int r     = lane & 15;             // row-in-tile (A) / column N (B,C)
    const int khalf = lane >> 4;             // 0: K={0,1}; 1: K={2,3}
    const int b0    = blockIdx.x * 16;
    const int row   = b0 + r;

    const float* prow = pred    + (size_t)row * T_TUN;
    const float* drow = demands + (size_t)row * D_DST;

    v8f c = {};
    // 1022 K-chunks of 4 tunnels, strided over the block's 4 waves.
    for (int ch = wave; ch < (T_TUN / 4); ch += 4) {
        const int t = ch * 4;
        // keep the per-row HBM stream ahead of us (global_prefetch_b8)
        __builtin_prefetch(prow + t + 256, 0, 1);

        // tunnel_to_link[t..t+3]: uniform address across the wave
        const int4 m = *reinterpret_cast<const int4*>(ttl + t);
        const int l0 = khalf ? m.z : m.x;
        const int l1 = khalf ? m.w : m.y;
        v2f bm;
        bm.x = (l0 == r) ? 1.0f : 0.0f;
        bm.y = (l1 == r) ? 1.0f : 0.0f;

        // A fragment: tunnel_traffic for K = tt, tt+1 (tt even -> same demand)
        const int    tt = t + 2 * khalf;
        const float2 p2 = *reinterpret_cast<const float2*>(prow + tt);
        const float  d  = drow[tt >> 3];
        v2f a;
        a.x = p2.x * d;
        a.y = p2.y * d;

        // D = A x B + C   (exact: B is one-hot f32)
        c = __builtin_amdgcn_wmma_f32_16x16x4_f32(
                /*neg_a=*/false, a, /*neg_b=*/false, bm,
                /*c_mod=*/(short)0, c, /*reuse_a=*/false, /*reuse_b=*/false);
    }

    // Fixed-order reduction of the 4 partial C tiles through LDS.
#pragma unroll
    for (int i = 0; i < 8; ++i) red[wave * 256 + lane * 8 + i] = c[i];
    __syncthreads();
    if (wave == 0) {
#pragma unroll
        for (int i = 0; i < 8; ++i) {
            const float s = red[0 * 256 + lane * 8 + i]
                          + red[1 * 256 + lane * 8 + i]
                          + red[2 * 256 + lane * 8 + i]
                          + red[3 * 256 + lane * 8 + i];
            const int mrow = b0 + i + 8 * khalf;       // C layout (see above)
            link_traffic[(size_t)mrow * L_LINKS + r] = s;
        }
    }
}

// ---------------------------------------------------------------------------
// Stage 2: per-row statistics + deterministic block reduction.
// One thread per row; 64 blocks x 256 threads cover B=16384 exactly.
// ---------------------------------------------------------------------------
__global__ __launch_bounds__(256)
void nal_stage2_rowstats(const float* __restrict__ link_traffic,
                         const float* __restrict__ neighbor_loads,
                         const float* __restrict__ caps,
                         float*       __restrict__ partials) {
    __shared__ float sdata[256];
    const int b = blockIdx.x * 256 + threadIdx.x;

    const float* lt = link_traffic  + (size_t)b * L_LINKS;
    const float* nl = neighbor_loads + (size_t)b * L_LINKS;

    float ltv[L_LINKS], nlv[L_LINKS], u[L_LINKS];
    float sum_u = 0.0f, sum_lt = 0.0f, sum_nl = 0.0f, mx = -3.0e38f;
#pragma unroll
    for (int l = 0; l < L_LINKS; ++l) {
        ltv[l] = lt[l];
        nlv[l] = nl[l];
        u[l]   = ltv[l] / (caps[l] + EPS);
        sum_u += u[l];
        sum_lt += ltv[l];
        sum_nl += nlv[l];
        mx = fmaxf(mx, u[l]);
    }
    const float mean = sum_u * (1.0f / 16.0f);
    float var = 0.0f;
#pragma unroll
    for (int l = 0; l < L_LINKS; ++l) { const float dv = u[l] - mean; var += dv * dv; }
    var *= (1.0f / 15.0f);                    // ddof = 1

    const float inv_lt = 1.0f / (sum_lt + EPS);
    const float inv_nl = 1.0f / (sum_nl + EPS);
    float cong = 0.0f;
#pragma unroll
    for (int l = 0; l < L_LINKS; ++l) cong += (ltv[l] * inv_lt) * (nlv[l] * inv_nl);

    sdata[threadIdx.x] = var + 0.5f * mx + 0.3f * cong;
    __syncthreads();
    for (int s = 128; s > 0; s >>= 1) {
        if (threadIdx.x < s) sdata[threadIdx.x] += sdata[threadIdx.x + s];
        __syncthreads();
    }
    if (threadIdx.x == 0) partials[blockIdx.x] = sdata[0];
}

// ---------------------------------------------------------------------------
// Stage 3: deterministic serial sum of 64 partials, scale by 1/B.
// ---------------------------------------------------------------------------
__global__ void nal_stage3_final(const float* __restrict__ partials,
                                 float* __restrict__ out, int n) {
    if (threadIdx.x == 0 && blockIdx.x == 0) {
        float s = 0.0f;
        for (int i = 0; i < n; ++i) s += partials[i];
        out[0] = s * (1.0f / (float)B_ROWS);
    }
}

extern "C" void kernel_launch(void* const* d_in, const int* in_sizes, int n_in,
                              void* d_out, int out_size, void* d_ws, size_t ws_size,
                              hipStream_t stream) {
    const float* pred     = (const float*)d_in[0];  // (B,T)
    const float* demands  = (const float*)d_in[1];  // (B,D)
    const float* neighbor = (const float*)d_in[2];  // (B,L)
    const float* caps     = (const float*)d_in[3];  // (L,)
    const int*   ttl      = (const int*)  d_in[4];  // (T,) tunnel_to_link
    // d_in[5] = dst_of_tunnel, recomputed analytically as t >> 3

    float* link_traffic = (float*)d_ws;                         // B*16 f32 = 1 MB
    float* partials     = link_traffic + (size_t)B_ROWS * L_LINKS; // +64 f32

    hipLaunchKernelGGL(nal_stage1_link_traffic,
                       dim3(B_ROWS / 16), dim3(128), 0, stream,
                       pred, demands, ttl, link_traffic);
    hipLaunchKernelGGL(nal_stage2_rowstats,
                       dim3(B_ROWS / 256), dim3(256), 0, stream,
                       link_traffic, neighbor, caps, partials);
    hipLaunchKernelGGL(nal_stage3_final,
                       dim3(1), dim3(32), 0, stream,
                       partials, (float*)d_out, B_ROWS / 256);
}